// Net_MP_46849503265406
// MI455X (gfx1250) — compile-verified
//
#include <hip/hip_runtime.h>
#include <hip/hip_bf16.h>
#include <stddef.h>

typedef __attribute__((ext_vector_type(2))) float v2f;
typedef __attribute__((ext_vector_type(8))) float v8f;

// ---------------------------------------------------------------------------
// Zero fill
// ---------------------------------------------------------------------------
__global__ void zero_f32(float* __restrict__ p, size_t n) {
    size_t i = (size_t)blockIdx.x * blockDim.x + threadIdx.x;
    if (i < n) p[i] = 0.0f;
}

// ---------------------------------------------------------------------------
// Build Wcat [32 x 160] row-major:
//   cols   0.. 95 : Wn2_d  (Wn2[d*1024 + i*32 + o])   d = col/32
//   cols  96..127 : Bn2    (bn2[i*32 + o])
//   cols 128..159 : root2  (root2[i*32 + o])
// ---------------------------------------------------------------------------
__global__ void prep_wcat(const float* __restrict__ Wn2, const float* __restrict__ bn2,
                          const float* __restrict__ root2, float* __restrict__ Wcat) {
    int t = blockIdx.x * blockDim.x + threadIdx.x;
    if (t >= 32 * 160) return;
    int i = t / 160, c = t % 160;
    float v;
    if (c < 96) {
        int d = c >> 5, o = c & 31;
        v = Wn2[d * 1024 + i * 32 + o];
    } else if (c < 128) {
        v = bn2[i * 32 + (c - 96)];
    } else {
        v = root2[i * 32 + (c - 128)];
    }
    Wcat[t] = v;
}

// ---------------------------------------------------------------------------
// conv1 node precompute:  Z[n, d*32+o] = sum_i x[n,i] * Wn1[d, i*32+o]   (d<3)
//                         Z[n, 96 +o ] = sum_i x[n,i] * bn1[i*32+o]
// (K=3 -> plain VALU, WMMA not worthwhile)
// ---------------------------------------------------------------------------
__global__ void conv1_node_pre(const float* __restrict__ x, const float* __restrict__ Wn1,
                               const float* __restrict__ bn1, float* __restrict__ Z, int N) {
    long long t = (long long)blockIdx.x * blockDim.x + threadIdx.x;
    if (t >= (long long)N * 128) return;
    int n = (int)(t >> 7);
    int c = (int)(t & 127);
    int d = c >> 5, o = c & 31;
    float x0 = x[n * 3 + 0], x1 = x[n * 3 + 1], x2 = x[n * 3 + 2];
    float s;
    if (d < 3) {
        const float* w = Wn1 + d * 96;
        s = fmaf(x0, w[o], fmaf(x1, w[32 + o], x2 * w[64 + o]));
    } else {
        s = fmaf(x0, bn1[o], fmaf(x1, bn1[32 + o], x2 * bn1[64 + o]));
    }
    Z[(size_t)n * 128 + c] = s;
}

// ---------------------------------------------------------------------------
// Edge kernel (both conv layers): one wave32 per edge, lane = out channel.
//   m = ea0*T[src][0..31] + ea1*T[src][32..63] + ea2*T[src][64..95] + T[src][96..127]
//   atomicAdd(agg[dst*32 + lane], m)
// Gather table T has row stride `stride` (128 for conv1's Z, 160 for conv2's Y);
// tables are L2-resident (51.2MB / 64MB << 192MB L2).
// ---------------------------------------------------------------------------
__global__ void conv_edge(const int* __restrict__ ei, const float* __restrict__ ea,
                          const float* __restrict__ T, float* __restrict__ agg,
                          int E, int stride) {
    int e    = (int)(((long long)blockIdx.x * blockDim.x + threadIdx.x) >> 5);
    int lane = threadIdx.x & 31;
    if (e >= E) return;
    int src = ei[e];
    int dst = ei[E + e];
    float a0 = ea[e * 3 + 0], a1 = ea[e * 3 + 1], a2 = ea[e * 3 + 2];
    const float* t = T + (size_t)src * stride;
    float m = fmaf(a0, t[lane],
              fmaf(a1, t[32 + lane],
              fmaf(a2, t[64 + lane], t[96 + lane])));
    atomicAdd(agg + (size_t)dst * 32 + lane, m);
}

// ---------------------------------------------------------------------------
// conv1 node epilogue (in-place on agg): h = relu(agg + x@root1 + b1)
// ---------------------------------------------------------------------------
__global__ void conv1_node_post(const float* __restrict__ x, const float* __restrict__ root1,
                                const float* __restrict__ b1, float* __restrict__ h, int N) {
    int n = (int)(((long long)blockIdx.x * blockDim.x + threadIdx.x) >> 5);
    int o = threadIdx.x & 31;
    if (n >= N) return;
    size_t idx = (size_t)n * 32 + o;
    float s = h[idx]
            + x[n * 3 + 0] * root1[o]
            + x[n * 3 + 1] * root1[32 + o]
            + x[n * 3 + 2] * root1[64 + o]
            + b1[o];
    h[idx] = fmaxf(s, 0.0f);
}

// ---------------------------------------------------------------------------
// conv2 node epilogue (in-place on agg): h2 = relu(agg2 + Y[:,128:160] + b2)
// (Y[:,128:160] = h1 @ root2, folded into the big WMMA GEMM)
// ---------------------------------------------------------------------------
__global__ void conv2_node_post(const float* __restrict__ Y, const float* __restrict__ b2,
                                float* __restrict__ h, int N) {
    int n = (int)(((long long)blockIdx.x * blockDim.x + threadIdx.x) >> 5);
    int o = threadIdx.x & 31;
    if (n >= N) return;
    size_t idx = (size_t)n * 32 + o;
    float s = h[idx] + Y[(size_t)n * 160 + 128 + o] + b2[o];
    h[idx] = fmaxf(s, 0.0f);
}

// ---------------------------------------------------------------------------
// WMMA fp32 GEMM: Out[nrows x NCOLS] = A[nrows x 32] @ W[32 x NCOLS]
// One wave computes a 16-row stripe. Uses V_WMMA_F32_16X16X4_F32, K in 8 steps.
// ISA VGPR layout (16x4 fp32 A, 4x16 fp32 B):
//   lane L, vgpr v:  A -> (row = L%16, K = kbase + 2*(L/16) + v)
//                    B -> (K  = kbase + 2*(L/16) + v, col = ntile*16 + L%16)
// EXEC stays all-ones for every WMMA (guards are wave-uniform or store-only).
// Store epilogue: wave-uniform full-tile fast path (8 unguarded stores with
// constant immediate offsets from one base pointer) + guarded cold tail.
// ---------------------------------------------------------------------------
template <int NCOLS>
__global__ void gemm32_wmma(const float* __restrict__ A, const float* __restrict__ W,
                            float* __restrict__ Out, int nrows) {
    int wave = (int)(((long long)blockIdx.x * blockDim.x + threadIdx.x) >> 5);
    int lane = threadIdx.x & 31;
    int rowBase = wave * 16;
    if (rowBase >= nrows) return;      // wave-uniform exit
    int half = lane >> 4;              // 0 or 1
    int l16  = lane & 15;
    int arow = rowBase + l16;
    if (arow >= nrows) arow = nrows - 1;   // clamp loads; stores are guarded
    bool fullTile = (rowBase + 16 <= nrows);   // wave-uniform

    v2f a[8];
#pragma unroll
    for (int kk = 0; kk < 8; ++kk) {
        const float* p = A + (size_t)arow * 32 + kk * 4 + half * 2;
        a[kk].x = p[0];
        a[kk].y = p[1];
    }

#pragma unroll
    for (int nt = 0; nt < NCOLS / 16; ++nt) {
        v8f c = {};
#pragma unroll
        for (int kk = 0; kk < 8; ++kk) {
            int k0 = kk * 4 + half * 2;
            v2f b;
            b.x = W[(k0 + 0) * NCOLS + nt * 16 + l16];
            b.y = W[(k0 + 1) * NCOLS + nt * 16 + l16];
            c = __builtin_amdgcn_wmma_f32_16x16x4_f32(
                    /*neg_a=*/false, a[kk], /*neg_b=*/false, b,
                    /*c_mod=*/(short)0, c, /*reuse_a=*/false, /*reuse_b=*/false);
        }
        float* outp = Out + (size_t)(rowBase + half * 8) * NCOLS + nt * 16 + l16;
        if (fullTile) {
            // hot path: constant-offset stores, no EXEC churn
#pragma unroll
            for (int v = 0; v < 8; ++v)
                outp[(size_t)v * NCOLS] = c[v];
        } else {
            // cold tail: per-row guard
#pragma unroll
            for (int v = 0; v < 8; ++v) {
                int row = rowBase + v + half * 8;
                if (row < nrows)
                    outp[(size_t)v * NCOLS] = c[v];
            }
        }
    }
}

// ---------------------------------------------------------------------------
// Final FC: out[n] = bf2 + sum_o relu(G[n,o] + bf1[o]) * Wf2[o]
// One wave per node, butterfly reduce across the wave32.
// ---------------------------------------------------------------------------
__global__ void fc_final(const float* __restrict__ G, const float* __restrict__ bf1,
                         const float* __restrict__ Wf2, const float* __restrict__ bf2,
                         float* __restrict__ out, int N) {
    int n = (int)(((long long)blockIdx.x * blockDim.x + threadIdx.x) >> 5);
    int o = threadIdx.x & 31;
    if (n >= N) return;
    float t = fmaxf(G[(size_t)n * 32 + o] + bf1[o], 0.0f) * Wf2[o];
#pragma unroll
    for (int m = 16; m >= 1; m >>= 1) t += __shfl_xor(t, m, 32);
    if (o == 0) out[n] = t + bf2[0];
}

// ---------------------------------------------------------------------------
// Launch
// ---------------------------------------------------------------------------
extern "C" void kernel_launch(void* const* d_in, const int* in_sizes, int n_in,
                              void* d_out, int out_size, void* d_ws, size_t ws_size,
                              hipStream_t stream) {
    const float* x     = (const float*)d_in[0];
    const int*   ei    = (const int*)  d_in[1];   // [2, E] (src row, dst row)
    const float* ea    = (const float*)d_in[2];   // [E, 3]
    const float* Wn1   = (const float*)d_in[3];   // [3, 96]
    const float* bn1   = (const float*)d_in[4];   // [96]
    const float* root1 = (const float*)d_in[5];   // [3, 32]
    const float* b1    = (const float*)d_in[6];   // [32]
    const float* Wn2   = (const float*)d_in[7];   // [3, 1024]
    const float* bn2   = (const float*)d_in[8];   // [1024]
    const float* root2 = (const float*)d_in[9];   // [32, 32]
    const float* b2    = (const float*)d_in[10];  // [32]
    const float* Wf1   = (const float*)d_in[11];  // [32, 32]
    const float* bf1   = (const float*)d_in[12];  // [32]
    const float* Wf2   = (const float*)d_in[13];  // [32, 1]
    const float* bf2   = (const float*)d_in[14];  // [1]
    float* out = (float*)d_out;

    const int N = in_sizes[0] / 3;   // 100000
    const int E = in_sizes[1] / 2;   // 400000

    // Workspace layout (floats):
    //   [0, 5120)              Wcat (32 x 160), padded to 32KB
    //   regA : N*32            agg1 -> h1 (in place) -> later G = h2 @ Wf1
    //   regB : N*32            agg2 -> h2 (in place)
    //   regZY: N*160           Z [N,128] for conv1, then Y [N,160] for conv2
    char*  ws    = (char*)d_ws;
    float* Wcat  = (float*)ws;
    float* regA  = (float*)(ws + 32768);
    float* regB  = regA + (size_t)N * 32;
    float* regZY = regB + (size_t)N * 32;

    const int BLK = 256;

    // 0) zero both agg regions (contiguous: N*64 floats)
    {
        size_t n = (size_t)N * 64;
        zero_f32<<<(unsigned)((n + BLK - 1) / BLK), BLK, 0, stream>>>(regA, n);
    }

    // 1) weight concat for conv2 GEMM
    prep_wcat<<<(32 * 160 + BLK - 1) / BLK, BLK, 0, stream>>>(Wn2, bn2, root2, Wcat);

    // 2) conv1: node precompute Z, edge gather/scatter, node epilogue
    {
        long long t = (long long)N * 128;
        conv1_node_pre<<<(unsigned)((t + BLK - 1) / BLK), BLK, 0, stream>>>(x, Wn1, bn1, regZY, N);
    }
    {
        long long t = (long long)E * 32;
        conv_edge<<<(unsigned)((t + BLK - 1) / BLK), BLK, 0, stream>>>(ei, ea, regZY, regA, E, 128);
    }
    {
        long long t = (long long)N * 32;
        conv1_node_post<<<(unsigned)((t + BLK - 1) / BLK), BLK, 0, stream>>>(x, root1, b1, regA, N);
    }

    // 3) conv2: Y = h1 @ [Wn2_0|Wn2_1|Wn2_2|Bn2|root2]  (WMMA), edges, epilogue
    {
        long long waves = (N + 15) / 16;
        long long t = waves * 32;
        gemm32_wmma<160><<<(unsigned)((t + BLK - 1) / BLK), BLK, 0, stream>>>(regA, Wcat, regZY, N);
    }
    {
        long long t = (long long)E * 32;
        conv_edge<<<(unsigned)((t + BLK - 1) / BLK), BLK, 0, stream>>>(ei, ea, regZY, regB, E, 160);
    }
    {
        long long t = (long long)N * 32;
        conv2_node_post<<<(unsigned)((t + BLK - 1) / BLK), BLK, 0, stream>>>(regZY, b2, regB, N);
    }

    // 4) FC head: G = h2 @ Wf1 (WMMA), then fused relu / Wf2 reduce
    {
        long long waves = (N + 15) / 16;
        long long t = waves * 32;
        gemm32_wmma<32><<<(unsigned)((t + BLK - 1) / BLK), BLK, 0, stream>>>(regB, Wf1, regA, N);
    }
    {
        long long t = (long long)N * 32;
        fc_final<<<(unsigned)((t + BLK - 1) / BLK), BLK, 0, stream>>>(regA, bf1, Wf2, bf2, out, N);
    }
}